// VolumetricRayMarcher_17171279250101
// MI455X (gfx1250) — compile-verified
//
#include <hip/hip_runtime.h>
#include <stdint.h>

// Volumetric ray marcher: B=4, R=65536, S=128.
// Memory-bound streaming scan: ~512 MB @ 23.3 TB/s => ~22 us floor.
// Strategy: TDM stages density tiles into LDS (with HW padding for
// conflict-free ds_load_b128), rgb streamed with non-temporal b128 loads,
// thread-per-ray sequential transmittance scan with exp2 fast path.

#define NUM_S  128
#define RPB    64          // rays per block == threads per block (2 waves)
#define ROW_DW 132         // 128 data dwords + 4 pad dwords per LDS row (528 B)

typedef float    v4f __attribute__((ext_vector_type(4)));
typedef uint32_t v4u __attribute__((ext_vector_type(4)));
typedef int32_t  v4i __attribute__((ext_vector_type(4)));
typedef int32_t  v8i __attribute__((ext_vector_type(8)));

__global__ __launch_bounds__(RPB)
void VolumetricRayMarcher_17171279250101_kernel(const float* __restrict__ rgb,
                                                const float* __restrict__ density,
                                                float* __restrict__ out)
{
    __shared__ float sdens[RPB * ROW_DW];   // 33,792 bytes

    const int      tid = threadIdx.x;
    const uint64_t ray = (uint64_t)blockIdx.x * RPB + tid;

#if defined(__gfx1250__) && __has_builtin(__builtin_amdgcn_tensor_load_to_lds)
    // ---- Tensor Data Mover: DMA 64x128 f32 density tile -> padded LDS ----
    if (tid < 32) {   // wave 0 issues the (per-wave) TDM op
        const uint64_t gaddr =
            (uint64_t)(uintptr_t)(density + (uint64_t)blockIdx.x * RPB * NUM_S);
        const uint32_t laddr = (uint32_t)(uintptr_t)(void*)sdens; // low 32b = LDS offset

        // D# group 0 (128b): count=1 | lds_addr | global_addr[56:0] | type=2
        v4u g0;
        g0.x = 1u;                                   // count=1, user descriptor
        g0.y = laddr;                                // lds_addr (bytes)
        g0.z = (uint32_t)gaddr;                      // global_addr[31:0]
        g0.w = (uint32_t)((gaddr >> 32) & 0x01FFFFFFu) | 0x80000000u; // [56:32] | type=2

        // D# group 1 (256b):
        //  d0: wg_mask=0 | data_size=2 (4B) | pad_enable | pad_interval=6 (128 dw)
        //      | pad_amount=3 (4 dw)
        //  tensor_dim0=128, tensor_dim1=64, tile_dim0=128, tile_dim1=64,
        //  tile_dim2=0, tensor_dim0_stride=128, tensor_dim1_stride=0
        v8i g1;
        g1[0] = (int32_t)((2u << 16) | (1u << 20) | (6u << 22) | (3u << 25));
        g1[1] = (int32_t)((uint32_t)NUM_S << 16);    // tensor_dim0[15:0] in [31:16]
        g1[2] = (int32_t)((uint32_t)RPB   << 16);    // dim0 hi=0 | tensor_dim1[15:0]
        g1[3] = (int32_t)((uint32_t)NUM_S << 16);    // dim1 hi=0 | tile_dim0=128
        g1[4] = (int32_t)RPB;                        // tile_dim1=64 | tile_dim2=0
        g1[5] = NUM_S;                               // tensor_dim0_stride lo
        g1[6] = 0;                                   // stride hi | dim1_stride lo
        g1[7] = 0;

        v4i g2 = {0, 0, 0, 0};                       // dims 2/3 unused (zeros legal)
        v4i g3 = {0, 0, 0, 0};
        v8i gx = {0, 0, 0, 0, 0, 0, 0, 0};           // extra group (6-arg toolchain), unused

        __builtin_amdgcn_tensor_load_to_lds(g0, g1, g2, g3, gx, /*cpol=*/0);
        __builtin_amdgcn_s_wait_tensorcnt(0);
    }
    __syncthreads();
#else
    // Fallback: plain fill of the padded LDS tile
    #pragma unroll 4
    for (int s4 = 0; s4 < NUM_S; s4 += 4)
        *(v4f*)&sdens[tid * ROW_DW + s4] =
            __builtin_nontemporal_load((const v4f*)(density + ray * NUM_S + s4));
    __syncthreads();
#endif

    // ---- per-ray transmittance scan ----
    const float NEG_DLOG2E = -(1.9f / 127.0f) * 1.4426950408889634f; // -delta*log2(e)
    const float LAST_M     = -1.0e10f * 1.4426950408889634f;         // last delta = 1e10

    float T = 1.0f, ax = 0.0f, ay = 0.0f, az = 0.0f;
    const v4f* rgb4  = (const v4f*)(rgb + ray * (uint64_t)(NUM_S * 3));
    const int  sbase = tid * ROW_DW;

    auto step = [&](float d, float r, float g, float b, float m) {
        float e  = __builtin_exp2f(d * m);      // exp(-d*delta) via v_exp_f32
        float te = T * e;
        float w  = T - te;                      // alpha * T_exclusive
        ax = __builtin_fmaf(w, r, ax);
        ay = __builtin_fmaf(w, g, ay);
        az = __builtin_fmaf(w, b, az);
        T  = __builtin_fmaf(T, 1e-10f, te);     // T *= (e + 1e-10)
    };

    auto group = [&](int i, float mlast) {
        v4f d4 = *(const v4f*)&sdens[sbase + 4 * i];           // ds_load_b128
        v4f c0 = __builtin_nontemporal_load(&rgb4[3 * i + 0]); // 48 B of rgb
        v4f c1 = __builtin_nontemporal_load(&rgb4[3 * i + 1]);
        v4f c2 = __builtin_nontemporal_load(&rgb4[3 * i + 2]);
        step(d4.x, c0.x, c0.y, c0.z, NEG_DLOG2E);
        step(d4.y, c0.w, c1.x, c1.y, NEG_DLOG2E);
        step(d4.z, c1.z, c1.w, c2.x, NEG_DLOG2E);
        step(d4.w, c2.y, c2.z, c2.w, mlast);
    };

    #pragma unroll 4
    for (int i = 0; i < 31; ++i) group(i, NEG_DLOG2E);
    group(31, LAST_M);                          // peel: last sample uses delta=1e10

    out[ray * 3 + 0] = ax;
    out[ray * 3 + 1] = ay;
    out[ray * 3 + 2] = az;
}

extern "C" void kernel_launch(void* const* d_in, const int* in_sizes, int n_in,
                              void* d_out, int out_size, void* d_ws, size_t ws_size,
                              hipStream_t stream) {
    const float* rgb     = (const float*)d_in[0];   // [B, R, S, 3] f32
    const float* density = (const float*)d_in[1];   // [B, R, S]    f32
    float*       out     = (float*)d_out;           // [B, R, 3]    f32

    const int total_rays = in_sizes[1] / NUM_S;     // B*R = 262144
    dim3 grid(total_rays / RPB);                    // 4096 blocks
    dim3 block(RPB);                                // 64 threads = 2 waves
    VolumetricRayMarcher_17171279250101_kernel<<<grid, block, 0, stream>>>(rgb, density, out);
}